// MultiHeadGraphAttention_59399397704166
// MI455X (gfx1250) — compile-verified
//
#include <hip/hip_runtime.h>
#include <hip/hip_bf16.h>

#define N_NODES 50000
#define N_EDGES 800000
#define D_IN    128
#define HEADS   8
#define UNITS   16
#define COLS    (HEADS * UNITS)   /* 128 */

typedef __attribute__((ext_vector_type(2))) float v2f;
typedef __attribute__((ext_vector_type(8))) float v8f;

// ---------------------------------------------------------------------------
// Monotone encoding of float -> unsigned so unsigned atomicMax == float max.
// encode key 0 decodes below every real float, so 0-init acts as -inf.
// ---------------------------------------------------------------------------
__device__ __forceinline__ unsigned encodeF(float f) {
    unsigned u = __float_as_uint(f);
    return (u & 0x80000000u) ? ~u : (u | 0x80000000u);
}
__device__ __forceinline__ float decodeF(unsigned u) {
    unsigned b = (u & 0x80000000u) ? (u ^ 0x80000000u) : ~u;
    return __uint_as_float(b);
}

// ---------------------------------------------------------------------------
// Kernel 0: zero output accumulator, softmax max-keys and sums
// ---------------------------------------------------------------------------
__global__ void __launch_bounds__(256)
gat_init(float* __restrict__ out, unsigned* __restrict__ smaxk,
         float* __restrict__ ssum) {
    int i = blockIdx.x * 256 + threadIdx.x;
    if (i < N_NODES * COLS) out[i] = 0.0f;
    if (i < N_NODES * HEADS) { smaxk[i] = 0u; ssum[i] = 0.0f; }
}

// ---------------------------------------------------------------------------
// Kernel 1: xp[n, h*16+u] = sum_d x[n,d] * W[d, h*16+u]   (fp32 WMMA)
// Block = 256 threads (8 waves), 128 node rows per block, W (64 KB) in LDS.
// Each wave: 16 rows x 128 cols via V_WMMA_F32_16X16X4_F32 over K=128.
// ---------------------------------------------------------------------------
__global__ void __launch_bounds__(256)
gat_gemm_wmma(const float* __restrict__ x, const float* __restrict__ W,
              float* __restrict__ xp) {
    __shared__ float Wlds[D_IN * COLS];            // 64 KB (WGP has 320 KB)
    for (int i = threadIdx.x; i < D_IN * COLS; i += 256)
        Wlds[i] = W[i];
    __syncthreads();

    const int wave = threadIdx.x >> 5;
    const int lane = threadIdx.x & 31;
    const int half = lane >> 4;                    // 0: lanes 0-15, 1: 16-31
    const int l16  = lane & 15;
    const int row0 = blockIdx.x * 128 + wave * 16;

    // A-fragment row (clamped so partial tiles keep EXEC all-ones for WMMA)
    int arow = row0 + l16;
    if (arow > N_NODES - 1) arow = N_NODES - 1;
    const float* __restrict__ Arow = x + (size_t)arow * D_IN;
    const int koff = half * 2;                     // K=0,1 low lanes; 2,3 high

    v8f acc[8] = {};
    for (int k = 0; k < D_IN; k += 4) {
        v2f a;
        a.x = Arow[k + koff];
        a.y = Arow[k + koff + 1];
#pragma unroll
        for (int ct = 0; ct < 8; ++ct) {
            const int n = ct * 16 + l16;
            v2f b;
            b.x = Wlds[(k + koff) * COLS + n];
            b.y = Wlds[(k + koff + 1) * COLS + n];
            acc[ct] = __builtin_amdgcn_wmma_f32_16x16x4_f32(
                false, a, false, b, (short)0, acc[ct], false, false);
        }
    }

    // C/D layout: VGPR v, lanes 0-15 -> M=v, lanes 16-31 -> M=v+8, N=lane&15
    const int mbase = row0 + half * 8;
    float* __restrict__ obase = xp + (size_t)mbase * COLS + l16;
    if (row0 + 16 <= N_NODES) {
        // Fast path: whole tile in range -> unguarded stores, one base pointer,
        // constant offsets (rows stride COLS floats, col tiles stride 16).
#pragma unroll
        for (int v = 0; v < 8; ++v) {
#pragma unroll
            for (int ct = 0; ct < 8; ++ct)
                obase[(size_t)v * COLS + ct * 16] = acc[ct][v];
        }
    } else {
        // Tail path (last block only)
#pragma unroll
        for (int v = 0; v < 8; ++v) {
            if (mbase + v < N_NODES) {
#pragma unroll
                for (int ct = 0; ct < 8; ++ct)
                    obase[(size_t)v * COLS + ct * 16] = acc[ct][v];
            }
        }
    }
}

// ---------------------------------------------------------------------------
// Kernel 2: per-(edge,head) attention logit + segment max (atomic, in L2)
// score[e,h] = sum_u leaky_relu(xp[t]+xp[s]+2*ba, 0.2)[h,u] * ka[h,u]
// ---------------------------------------------------------------------------
__global__ void __launch_bounds__(256)
gat_edge_score(const float* __restrict__ xp, const int* __restrict__ edges,
               const float* __restrict__ ka, const float* __restrict__ ba,
               float* __restrict__ scores, unsigned* __restrict__ smaxk) {
    int idx = blockIdx.x * 256 + threadIdx.x;
    if (idx >= N_EDGES * HEADS) return;
    const int e = idx >> 3;
    const int h = idx & 7;
    const int s = edges[2 * e + 0];
    const int t = edges[2 * e + 1];

    const float4* __restrict__ pt = (const float4*)(xp + (size_t)t * COLS + h * UNITS);
    const float4* __restrict__ ps = (const float4*)(xp + (size_t)s * COLS + h * UNITS);
    const float4* __restrict__ pa = (const float4*)(ka + h * UNITS);
    const float4* __restrict__ pb = (const float4*)(ba + h * UNITS);

    float sc = 0.0f;
#pragma unroll
    for (int q = 0; q < 4; ++q) {
        float4 vt = pt[q], vs = ps[q], va = pa[q], vb = pb[q];
        float v;
        v = vt.x + vs.x + 2.0f * vb.x; v = (v > 0.0f) ? v : 0.2f * v; sc += v * va.x;
        v = vt.y + vs.y + 2.0f * vb.y; v = (v > 0.0f) ? v : 0.2f * v; sc += v * va.y;
        v = vt.z + vs.z + 2.0f * vb.z; v = (v > 0.0f) ? v : 0.2f * v; sc += v * va.z;
        v = vt.w + vs.w + 2.0f * vb.w; v = (v > 0.0f) ? v : 0.2f * v; sc += v * va.w;
    }
    scores[idx] = sc;
    atomicMax(smaxk + (size_t)t * HEADS + h, encodeF(sc));
}

// ---------------------------------------------------------------------------
// Kernel 3: p = exp(score - segmax[t]); segment sum of p
// ---------------------------------------------------------------------------
__global__ void __launch_bounds__(256)
gat_edge_softmax(float* __restrict__ scores, const int* __restrict__ edges,
                 const unsigned* __restrict__ smaxk, float* __restrict__ ssum) {
    int idx = blockIdx.x * 256 + threadIdx.x;
    if (idx >= N_EDGES * HEADS) return;
    const int e = idx >> 3;
    const int h = idx & 7;
    const int t = edges[2 * e + 1];
    const float m = decodeF(smaxk[(size_t)t * HEADS + h]);
    const float p = expf(scores[idx] - m);
    scores[idx] = p;
    atomicAdd(ssum + (size_t)t * HEADS + h, p);
}

// ---------------------------------------------------------------------------
// Kernel 4: out[t,h,:] += (p / (ssum[t,h]+1e-7)) * xp[s,h,:]
// ---------------------------------------------------------------------------
__global__ void __launch_bounds__(256)
gat_edge_aggregate(const float* __restrict__ scores, const int* __restrict__ edges,
                   const float* __restrict__ ssum, const float* __restrict__ xp,
                   float* __restrict__ out) {
    int idx = blockIdx.x * 256 + threadIdx.x;
    if (idx >= N_EDGES * HEADS) return;
    const int e = idx >> 3;
    const int h = idx & 7;
    const int s = edges[2 * e + 0];
    const int t = edges[2 * e + 1];
    const float w = scores[idx] / (ssum[(size_t)t * HEADS + h] + 1e-7f);
    const float* __restrict__ xs = xp + (size_t)s * COLS + h * UNITS;
    float* __restrict__ o = out + (size_t)t * COLS + h * UNITS;
#pragma unroll
    for (int u = 0; u < UNITS; ++u)
        atomicAdd(o + u, w * xs[u]);
}

// ---------------------------------------------------------------------------
// Kernel 5: out = gelu_tanh(out + bias), in place on d_out
// ---------------------------------------------------------------------------
__global__ void __launch_bounds__(256)
gat_finalize(float* __restrict__ out, const float* __restrict__ bias) {
    int i = blockIdx.x * 256 + threadIdx.x;
    if (i >= N_NODES * COLS) return;
    const int c = i & (COLS - 1);
    float v = out[i] + bias[c];
    const float k0 = 0.7978845608028654f;   // sqrt(2/pi)
    float g = 0.5f * v * (1.0f + tanhf(k0 * (v + 0.044715f * v * v * v)));
    out[i] = g;
}

// ---------------------------------------------------------------------------
extern "C" void kernel_launch(void* const* d_in, const int* in_sizes, int n_in,
                              void* d_out, int out_size, void* d_ws, size_t ws_size,
                              hipStream_t stream) {
    const float* x     = (const float*)d_in[0];
    const int*   edges = (const int*)d_in[1];
    const float* W     = (const float*)d_in[2];   // (128, 8, 16) row-major = 128x128
    const float* ka    = (const float*)d_in[3];   // (1, 8, 16)
    const float* ba    = (const float*)d_in[4];   // (1, 8, 16)
    const float* bias  = (const float*)d_in[5];   // (128,)
    float* out = (float*)d_out;

    // workspace carve-up (floats): xp | scores | smax keys | ssum  (~54 MB)
    float*    xp     = (float*)d_ws;
    float*    scores = xp + (size_t)N_NODES * COLS;
    unsigned* smaxk  = (unsigned*)(scores + (size_t)N_EDGES * HEADS);
    float*    ssum   = (float*)(smaxk + (size_t)N_NODES * HEADS);

    const int nc_blocks = (N_NODES * COLS + 255) / 256;     // 25000
    const int eh_blocks = (N_EDGES * HEADS + 255) / 256;    // 25000
    const int gemm_blocks = (N_NODES + 127) / 128;          // 391

    gat_init<<<nc_blocks, 256, 0, stream>>>(out, smaxk, ssum);
    gat_gemm_wmma<<<gemm_blocks, 256, 0, stream>>>(x, W, xp);
    gat_edge_score<<<eh_blocks, 256, 0, stream>>>(xp, edges, ka, ba, scores, smaxk);
    gat_edge_softmax<<<eh_blocks, 256, 0, stream>>>(scores, edges, smaxk, ssum);
    gat_edge_aggregate<<<eh_blocks, 256, 0, stream>>>(scores, edges, ssum, xp, out);
    gat_finalize<<<nc_blocks, 256, 0, stream>>>(out, bias);
}